// GCN_74397423501365
// MI455X (gfx1250) — compile-verified
//
#include <hip/hip_runtime.h>
#include <hip/hip_bf16.h>

typedef float v2f __attribute__((ext_vector_type(2)));
typedef float v8f __attribute__((ext_vector_type(8)));

#define N_NODES        140000
#define N_EDGES        2240000
#define NFEAT          128
#define NHID           128
#define N_FRAMES       14
#define NODES_PER_FRM  10000
#define EDGES_PER_WAVE 64

// --------------------------------------------------------------------------
// Kernel 1: support = x @ W   (fp32 WMMA 16x16x4, W staged in LDS)
// Block = 256 threads = 8 waves. Wave w owns N-tile [16w,16w+16); block owns
// 32 rows (2 M-tiles of 16). Grid = 140000/32 = 4375 (exact).
// fp32 A 16x4 layout (ISA 7.12.2): lane holds M=lane&15; VGPR r holds
// K = k + 2*(lane>>4) + r. B 4x16 symmetric. C/D: VGPR r -> M = r + 8*(lane>>4).
// --------------------------------------------------------------------------
__global__ __launch_bounds__(256) void gcn_gemm_xw(const float* __restrict__ x,
                                                   const float* __restrict__ W,
                                                   float* __restrict__ support) {
    __shared__ float sW[NFEAT * NHID];   // 64 KB of the 320 KB WGP LDS
    const int tid = threadIdx.x;
    {   // cooperative W load, float4
        const float4* src = (const float4*)W;
        float4*       dst = (float4*)sW;
        #pragma unroll
        for (int i = tid; i < (NFEAT * NHID) / 4; i += 256) dst[i] = src[i];
    }
    __syncthreads();

    const int lane  = tid & 31;
    const int wave  = tid >> 5;        // 0..7 -> N tile
    const int n0    = wave * 16;
    const int lhalf = lane >> 4;       // 0 or 1
    const int l16   = lane & 15;
    const int rowBase = blockIdx.x * 32;

    #pragma unroll
    for (int mt = 0; mt < 2; ++mt) {
        const int m0 = rowBase + mt * 16;
        const float* __restrict__ xrow = x + (long)(m0 + l16) * NFEAT;
        v8f c = {};
        #pragma unroll 4
        for (int k = 0; k < NFEAT; k += 4) {
            const int ka = k + lhalf * 2;
            v2f a, b;
            a[0] = xrow[ka];
            a[1] = xrow[ka + 1];
            b[0] = sW[(ka + 0) * NHID + n0 + l16];
            b[1] = sW[(ka + 1) * NHID + n0 + l16];
            c = __builtin_amdgcn_wmma_f32_16x16x4_f32(false, a, false, b,
                                                      (short)0, c, false, false);
        }
        #pragma unroll
        for (int r = 0; r < 8; ++r) {
            const int row = m0 + r + lhalf * 8;
            support[(long)row * NHID + n0 + l16] = c[r];
        }
    }
}

// --------------------------------------------------------------------------
// Kernel 2: zero-fill agg (float4 stores)
// --------------------------------------------------------------------------
__global__ __launch_bounds__(256) void gcn_zero_f4(float* __restrict__ p, long n4) {
    long i = (long)blockIdx.x * blockDim.x + threadIdx.x;
    if (i < n4) ((float4*)p)[i] = make_float4(0.f, 0.f, 0.f, 0.f);
}

// --------------------------------------------------------------------------
// Kernel 3: msgs + segment_sum. One wave per 64-edge chunk; lane owns 4 cols.
// edge_row is sorted -> accumulate in registers until the row changes, then
// flush with 4 f32 atomics. Chunk-boundary overlap handled by atomics.
// --------------------------------------------------------------------------
__global__ __launch_bounds__(256) void gcn_edge_agg(const float* __restrict__ support,
                                                    const float* __restrict__ eval,
                                                    const int*   __restrict__ erow,
                                                    const int*   __restrict__ ecol,
                                                    float*       __restrict__ agg,
                                                    int nEdges) {
    const int lane = threadIdx.x & 31;
    const int wv   = blockIdx.x * (blockDim.x >> 5) + (threadIdx.x >> 5);
    long e0 = (long)wv * EDGES_PER_WAVE;
    if (e0 >= nEdges) return;
    long e1 = e0 + EDGES_PER_WAVE;
    if (e1 > nEdges) e1 = nEdges;

    const int c4 = lane * 4;
    float4 acc = make_float4(0.f, 0.f, 0.f, 0.f);
    int curRow = erow[e0];

    for (long e = e0; e < e1; ++e) {
        const int r = erow[e];                 // uniform across the wave
        if (r != curRow) {
            float* dst = agg + (long)curRow * NHID + c4;
            atomicAdd(dst + 0, acc.x);
            atomicAdd(dst + 1, acc.y);
            atomicAdd(dst + 2, acc.z);
            atomicAdd(dst + 3, acc.w);
            acc = make_float4(0.f, 0.f, 0.f, 0.f);
            curRow = r;
        }
        const int   cIdx = ecol[e];
        const float v    = eval[e];
        const float4 s = *(const float4*)(support + (long)cIdx * NHID + c4);
        acc.x += v * s.x;  acc.y += v * s.y;
        acc.z += v * s.z;  acc.w += v * s.w;
    }
    float* dst = agg + (long)curRow * NHID + c4;
    atomicAdd(dst + 0, acc.x);
    atomicAdd(dst + 1, acc.y);
    atomicAdd(dst + 2, acc.z);
    atomicAdd(dst + 3, acc.w);
}

// --------------------------------------------------------------------------
// Kernel 4: h = relu(agg + b); frame-max over 14; row log_softmax.
// One wave per output node; lane owns 4 cols; wave32 shuffle reductions.
// --------------------------------------------------------------------------
__global__ __launch_bounds__(256) void gcn_head(const float* __restrict__ agg,
                                                const float* __restrict__ bias,
                                                float* __restrict__ out) {
    const int lane = threadIdx.x & 31;
    const int node = blockIdx.x * (blockDim.x >> 5) + (threadIdx.x >> 5);
    if (node >= NODES_PER_FRM) return;

    const int c4 = lane * 4;
    const float4 b4 = *(const float4*)(bias + c4);

    // relu output >= 0, all 14 frames present -> init 0 is the true max's floor
    float4 h = make_float4(0.f, 0.f, 0.f, 0.f);
    #pragma unroll
    for (int f = 0; f < N_FRAMES; ++f) {
        const float4 a = *(const float4*)(agg + ((long)f * NODES_PER_FRM + node) * NHID + c4);
        h.x = fmaxf(h.x, fmaxf(a.x + b4.x, 0.f));
        h.y = fmaxf(h.y, fmaxf(a.y + b4.y, 0.f));
        h.z = fmaxf(h.z, fmaxf(a.z + b4.z, 0.f));
        h.w = fmaxf(h.w, fmaxf(a.w + b4.w, 0.f));
    }

    // row max across 128 values (4 local + wave32 xor-shuffle tree)
    float m = fmaxf(fmaxf(h.x, h.y), fmaxf(h.z, h.w));
    #pragma unroll
    for (int off = 16; off > 0; off >>= 1) m = fmaxf(m, __shfl_xor(m, off, 32));

    float s = __expf(h.x - m) + __expf(h.y - m) + __expf(h.z - m) + __expf(h.w - m);
    #pragma unroll
    for (int off = 16; off > 0; off >>= 1) s += __shfl_xor(s, off, 32);

    const float lse = m + __logf(s);
    float4 o = make_float4(h.x - lse, h.y - lse, h.z - lse, h.w - lse);
    *(float4*)(out + (long)node * NHID + c4) = o;
}

// --------------------------------------------------------------------------
// Launch. Inputs: 0=x 1=W 2=b 3=edge_val 4=edge_row 5=edge_col.
// Workspace: support [140000*128 f32] then agg [140000*128 f32] (143.4 MB).
// --------------------------------------------------------------------------
extern "C" void kernel_launch(void* const* d_in, const int* in_sizes, int n_in,
                              void* d_out, int out_size, void* d_ws, size_t ws_size,
                              hipStream_t stream) {
    (void)in_sizes; (void)n_in; (void)out_size; (void)ws_size;
    const float* x    = (const float*)d_in[0];
    const float* W    = (const float*)d_in[1];
    const float* b    = (const float*)d_in[2];
    const float* ev   = (const float*)d_in[3];
    const int*   erow = (const int*)  d_in[4];
    const int*   ecol = (const int*)  d_in[5];
    float* out = (float*)d_out;

    float* support = (float*)d_ws;
    float* agg     = support + (long)N_NODES * NHID;   // 71,680,000 B offset (16B aligned)

    // 1) support = x @ W  (WMMA fp32)
    gcn_gemm_xw<<<N_NODES / 32, 256, 0, stream>>>(x, W, support);

    // 2) agg = 0
    const long n4 = (long)N_NODES * NHID / 4;          // 4,480,000 float4s
    gcn_zero_f4<<<(int)((n4 + 255) / 256), 256, 0, stream>>>(agg, n4);

    // 3) agg += segment_sum(edge_val * support[edge_col])
    const int nWaves  = (N_EDGES + EDGES_PER_WAVE - 1) / EDGES_PER_WAVE; // 35000
    const int nBlocks = (nWaves + 7) / 8;                                // 4375
    gcn_edge_agg<<<nBlocks, 256, 0, stream>>>(support, ev, erow, ecol, agg, N_EDGES);

    // 4) relu+bias, frame max, log_softmax
    gcn_head<<<(NODES_PER_FRM + 7) / 8, 256, 0, stream>>>(agg, b, out);
}